// Model_83580063580399
// MI455X (gfx1250) — compile-verified
//
#include <hip/hip_runtime.h>
#include <cstdint>

#define NUM_USERS 100000
#define NUM_ITEMS 200000
#define N_NODES   (NUM_USERS + NUM_ITEMS)
#define DIM       64
#define NNZ       4000000
#define EPB       2048          // edges per block (8 waves x 256)
#define EPW       256           // edges per wave slice (NNZ % EPW == 0)

static const size_t NELEM  = (size_t)N_NODES * DIM;   // 19,200,000 floats (76.8 MB)
static const size_t NELEM4 = NELEM / 4;               // 4,800,000 float4

// --- CDNA5 async global->LDS helpers ---------------------------------------
// Probe-confirmed: __builtin_amdgcn_global_load_async_to_lds_b128 exists,
// (v4i addrspace(1)*, v4i addrspace(3)*, imm offset, imm cpol).
typedef int v4i_t __attribute__((vector_size(16)));

#define AS1_V4IP(p) ((__attribute__((address_space(1))) v4i_t*)(uintptr_t)(p))
#define AS3_V4IP(p) ((__attribute__((address_space(3))) v4i_t*)(unsigned)(uintptr_t)(p))

__device__ __forceinline__ void async_g2l_b128(const void* g, void* l) {
#if __has_builtin(__builtin_amdgcn_global_load_async_to_lds_b128)
    __builtin_amdgcn_global_load_async_to_lds_b128(AS1_V4IP(g), AS3_V4IP(l), 0, 0);
#else
    *(float4*)l = *(const float4*)g;   // synchronous staging fallback
#endif
}

__device__ __forceinline__ void wait_async_0() {
#if __has_builtin(__builtin_amdgcn_s_wait_asynccnt)
    __builtin_amdgcn_s_wait_asynccnt(0);
#else
    asm volatile("s_wait_asynccnt 0x0" ::: "memory");
#endif
}

// ---------------------------------------------------------------------------
// scatter SpMM: one wave32 per 256-edge slice; the slice's row/col/val stream
// is staged into LDS via async-to-LDS (ASYNCcnt is per-wave, slices are
// wave-private, so no block barrier). Lanes cover the 64-dim row as float2;
// gather rows stay L2-resident (each phase touches exactly 2 x 76.8 MB
// tables < 192 MB L2); scatter via no-return global_atomic_add_f32.
// ---------------------------------------------------------------------------
__global__ __launch_bounds__(256) void lgcn_spmm_kernel(
    const int*   __restrict__ rows, const int* __restrict__ cols,
    const float* __restrict__ vals, const float* __restrict__ x,
    float*       __restrict__ y)
{
    __shared__ int   s_rows[EPB];
    __shared__ int   s_cols[EPB];
    __shared__ float s_vals[EPB];

    const int lane  = threadIdx.x & 31;          // wave32
    const int wave  = threadIdx.x >> 5;
    const int slice = wave * EPW;
    const long long sbeg = (long long)blockIdx.x * EPB + slice;
    if (sbeg >= NNZ) return;                     // NNZ % EPW == 0 -> full slices only

    {
        const int*   gr = rows + sbeg;
        const int*   gc = cols + sbeg;
        const float* gv = vals + sbeg;
        async_g2l_b128(gr +       lane * 4, &s_rows[slice +       lane * 4]);
        async_g2l_b128(gr + 128 + lane * 4, &s_rows[slice + 128 + lane * 4]);
        async_g2l_b128(gc +       lane * 4, &s_cols[slice +       lane * 4]);
        async_g2l_b128(gc + 128 + lane * 4, &s_cols[slice + 128 + lane * 4]);
        async_g2l_b128(gv +       lane * 4, &s_vals[slice +       lane * 4]);
        async_g2l_b128(gv + 128 + lane * 4, &s_vals[slice + 128 + lane * 4]);
        wait_async_0();
    }

    // 1-deep software pipeline: gather row for edge i+1 while edge i's
    // atomics issue (hides L2 gather latency inside the wave).
    int c0 = s_cols[slice];
    float2 xv = *(const float2*)(x + (size_t)c0 * DIM + lane * 2);

    #pragma unroll 4
    for (int i = 0; i < EPW; ++i) {
        const float2 cur = xv;
        if (i + 1 < EPW) {
            const int cn = s_cols[slice + i + 1];
            xv = *(const float2*)(x + (size_t)cn * DIM + lane * 2);
        }
        const int   r = s_rows[slice + i];
        const float v = s_vals[slice + i];
        float* dst = y + (size_t)r * DIM + lane * 2;
        __hip_atomic_fetch_add(dst,     v * cur.x, __ATOMIC_RELAXED, __HIP_MEMORY_SCOPE_AGENT);
        __hip_atomic_fetch_add(dst + 1, v * cur.y, __ATOMIC_RELAXED, __HIP_MEMORY_SCOPE_AGENT);
    }
}

// ---------------------------------------------------------------------------
// init: bufA = x0 = concat(user_emb, item_emb); bufB = 0 (layer-0 target);
//       acc(d_out) = 0 (layer-2 target; also re-zeros poisoned d_out per call)
// ---------------------------------------------------------------------------
__global__ __launch_bounds__(256) void lgcn_init_kernel(
    const float4* __restrict__ uemb, const float4* __restrict__ iemb,
    float4* __restrict__ x, float4* __restrict__ y, float4* __restrict__ acc)
{
    size_t i = (size_t)blockIdx.x * blockDim.x + threadIdx.x;
    if (i >= NELEM4) return;
    const size_t nu4 = (size_t)NUM_USERS * DIM / 4;
    float4 v = (i < nu4) ? uemb[i] : iemb[i - nu4];
    x[i] = v;
    float4 z; z.x = 0.f; z.y = 0.f; z.z = 0.f; z.w = 0.f;
    y[i]   = z;
    acc[i] = z;
}

// ---------------------------------------------------------------------------
// zero: clear a table (bufA between layers 0 and 1, once its x0 is dead)
// ---------------------------------------------------------------------------
__global__ __launch_bounds__(256) void lgcn_zero_kernel(float4* __restrict__ buf)
{
    size_t i = (size_t)blockIdx.x * blockDim.x + threadIdx.x;
    if (i >= NELEM4) return;
    float4 z; z.x = 0.f; z.y = 0.f; z.z = 0.f; z.w = 0.f;
    buf[i] = z;
}

// ---------------------------------------------------------------------------
// final fused mean: out = (x0(recomputed from inputs) + x1 + x2 + x3) * 0.25
//   x1 lives in bufB, x2 in bufA, x3 already in d_out.
// ---------------------------------------------------------------------------
__global__ __launch_bounds__(256) void lgcn_final_kernel(
    const float4* __restrict__ uemb, const float4* __restrict__ iemb,
    const float4* __restrict__ x1,   const float4* __restrict__ x2,
    float4* __restrict__ out)
{
    size_t i = (size_t)blockIdx.x * blockDim.x + threadIdx.x;
    if (i >= NELEM4) return;
    const size_t nu4 = (size_t)NUM_USERS * DIM / 4;
    float4 v0 = (i < nu4) ? uemb[i] : iemb[i - nu4];
    float4 a = x1[i];
    float4 b = x2[i];
    float4 c = out[i];
    c.x = (v0.x + a.x + b.x + c.x) * 0.25f;
    c.y = (v0.y + a.y + b.y + c.y) * 0.25f;
    c.z = (v0.z + a.z + b.z + c.z) * 0.25f;
    c.w = (v0.w + a.w + b.w + c.w) * 0.25f;
    out[i] = c;
}

extern "C" void kernel_launch(void* const* d_in, const int* in_sizes, int n_in,
                              void* d_out, int out_size, void* d_ws, size_t ws_size,
                              hipStream_t stream)
{
    const float* user_emb  = (const float*)d_in[0];  // 100000*64
    const float* item_emb  = (const float*)d_in[1];  // 200000*64
    const int*   edge_rows = (const int*)  d_in[2];  // 4M
    const int*   edge_cols = (const int*)  d_in[3];  // 4M
    const float* edge_vals = (const float*)d_in[4];  // 4M

    float* acc  = (float*)d_out;            // x3 target, then final output
    float* bufA = (float*)d_ws;             // 76.8 MB : x0, then x2
    float* bufB = bufA + NELEM;             // 76.8 MB : x1      (ws needs 153.6 MB)

    const int ethreads = 256;
    const dim3 eblk((unsigned)((NELEM4 + ethreads - 1) / ethreads)); // 18750
    const dim3 sblk((unsigned)((NNZ + EPB - 1) / EPB));              // 1954

    // bufA = x0, bufB = 0, d_out = 0
    lgcn_init_kernel<<<eblk, ethreads, 0, stream>>>(
        (const float4*)user_emb, (const float4*)item_emb,
        (float4*)bufA, (float4*)bufB, (float4*)acc);

    // layer 0: x1 = A @ x0         (read A, write B)
    lgcn_spmm_kernel<<<sblk, ethreads, 0, stream>>>(edge_rows, edge_cols, edge_vals, bufA, bufB);

    // x0 in bufA now dead -> zero it as layer-1 target
    lgcn_zero_kernel<<<eblk, ethreads, 0, stream>>>((float4*)bufA);

    // layer 1: x2 = A @ x1         (read B, write A)
    lgcn_spmm_kernel<<<sblk, ethreads, 0, stream>>>(edge_rows, edge_cols, edge_vals, bufB, bufA);

    // layer 2: x3 = A @ x2         (read A, write d_out)
    lgcn_spmm_kernel<<<sblk, ethreads, 0, stream>>>(edge_rows, edge_cols, edge_vals, bufA, acc);

    // out = (x0 + x1 + x2 + x3) / 4
    lgcn_final_kernel<<<eblk, ethreads, 0, stream>>>(
        (const float4*)user_emb, (const float4*)item_emb,
        (const float4*)bufB, (const float4*)bufA, (float4*)acc);
}